// GNN7L_Sage_6571299962947
// MI455X (gfx1250) — compile-verified
//
#include <hip/hip_runtime.h>
#include <hip/hip_bf16.h>

typedef __attribute__((ext_vector_type(2))) float v2f;
typedef __attribute__((ext_vector_type(8))) float v8f;

#define NEG_INF_BITS 0xFF800000u

// ---------------------------------------------------------------------------
// Fill agg buffer with -inf bit pattern (segment_max identity).
// ---------------------------------------------------------------------------
__global__ __launch_bounds__(256) void fill_neg_inf(unsigned int* __restrict__ p, int n) {
    int i = blockIdx.x * blockDim.x + threadIdx.x;
    if (i < n) p[i] = NEG_INF_BITS;
}

// ---------------------------------------------------------------------------
// Edge scatter-max: agg[dst][:] = max(agg[dst][:], x[src][:])
// Sign-split trick: positive floats -> signed atomicMax on int bits,
// negative floats -> unsigned atomicMin on bits. Consistent total order,
// native GLOBAL_ATOMIC_MAX_I32 / GLOBAL_ATOMIC_MIN_U32, no CAS loop.
// ---------------------------------------------------------------------------
__global__ __launch_bounds__(256) void sage_scatter_max(const int* __restrict__ src,
                                                        const int* __restrict__ dst,
                                                        const float* __restrict__ x,
                                                        float* __restrict__ agg, int E) {
    int e = blockIdx.x * blockDim.x + threadIdx.x;
    if (e >= E) return;
    // prefetch edge indices ahead (global_prefetch_b8)
    __builtin_prefetch(src + e + 2048, 0, 1);
    __builtin_prefetch(dst + e + 2048, 0, 1);
    int s = src[e];
    int d = dst[e];
    const float4* __restrict__ xs = (const float4*)(x + (size_t)s * 16);
    int* __restrict__ ag = (int*)(agg + (size_t)d * 16);
#pragma unroll
    for (int g = 0; g < 4; ++g) {
        float4 v = xs[g];
        float vals[4] = {v.x, v.y, v.z, v.w};
#pragma unroll
        for (int j = 0; j < 4; ++j) {
            int bits = __float_as_int(vals[j]);
            if (bits >= 0)
                atomicMax(ag + g * 4 + j, bits);
            else
                atomicMin((unsigned int*)(ag + g * 4 + j), (unsigned int)bits);
        }
    }
}

// ---------------------------------------------------------------------------
// Dense SAGE update for one layer of width 16:
//   out[n,:] = act( fix(agg[n,:]) @ Wl + x[n,:] @ Wr + b )
// One wave32 per 16-node tile; V_WMMA_F32_16X16X4_F32 x8 (4 K-steps, 2 mats).
// A layout (16x4 f32): lane L holds (M = L%16, K = 2*(L/16)+j) in vgpr j.
// B layout mirrors A with N = L%16. C/D: vgpr r -> M = r + 8*(L/16), N = L%16.
// ---------------------------------------------------------------------------
template <bool RELU>
__global__ __launch_bounds__(256) void sage_dense_wmma(const float* __restrict__ agg,
                                                       const float* __restrict__ xin,
                                                       const float* __restrict__ Wl,
                                                       const float* __restrict__ Wr,
                                                       const float* __restrict__ bias,
                                                       float* __restrict__ out, int nTiles) {
    int wave = (blockIdx.x * blockDim.x + threadIdx.x) >> 5;
    int lane = threadIdx.x & 31;
    if (wave >= nTiles) return;  // wave-uniform: active waves keep EXEC all-1s

    int col   = lane & 15;   // M index for A loads; N index for B / C / bias
    int khalf = lane >> 4;   // 0 or 1
    int base  = wave * 16;

    const float* __restrict__ aggRow = agg + (size_t)(base + col) * 16;
    const float* __restrict__ xRow   = xin + (size_t)(base + col) * 16;

    v8f c = {};
#pragma unroll
    for (int kb = 0; kb < 4; ++kb) {
        int k0 = kb * 4 + khalf * 2;
        v2f aA, aX, bL, bR;
        float a0 = aggRow[k0], a1 = aggRow[k0 + 1];
        // isolated-node fixup: -inf -> 0 (matches jnp.where(isfinite, agg, 0))
        aA.x = (__float_as_uint(a0) == NEG_INF_BITS) ? 0.0f : a0;
        aA.y = (__float_as_uint(a1) == NEG_INF_BITS) ? 0.0f : a1;
        aX.x = xRow[k0];
        aX.y = xRow[k0 + 1];
        bL.x = Wl[(k0) * 16 + col];
        bL.y = Wl[(k0 + 1) * 16 + col];
        bR.x = Wr[(k0) * 16 + col];
        bR.y = Wr[(k0 + 1) * 16 + col];
        c = __builtin_amdgcn_wmma_f32_16x16x4_f32(false, aA, false, bL, (short)0, c, false, false);
        c = __builtin_amdgcn_wmma_f32_16x16x4_f32(false, aX, false, bR, (short)0, c, false, false);
    }

    float bv = bias[col];
    int mbase = base + (khalf ? 8 : 0);
#pragma unroll
    for (int r = 0; r < 8; ++r) {
        float v = c[r] + bv;
        if (RELU) v = fmaxf(v, 0.0f);
        out[(size_t)(mbase + r) * 16 + col] = v;
    }
}

// ---------------------------------------------------------------------------
// Final layer (16 -> 2) + log_softmax over 2 classes, one thread per node.
// ---------------------------------------------------------------------------
__global__ __launch_bounds__(256) void sage_final(const float* __restrict__ agg,
                                                  const float* __restrict__ h,
                                                  const float* __restrict__ Wl7,
                                                  const float* __restrict__ Wr7,
                                                  const float* __restrict__ b7,
                                                  float* __restrict__ out, int N) {
    int n = blockIdx.x * blockDim.x + threadIdx.x;
    if (n >= N) return;
    float z0 = b7[0], z1 = b7[1];
    const float* ar = agg + (size_t)n * 16;
    const float* hr = h + (size_t)n * 16;
#pragma unroll
    for (int k = 0; k < 16; ++k) {
        float a = ar[k];
        a = (__float_as_uint(a) == NEG_INF_BITS) ? 0.0f : a;
        float hx = hr[k];
        z0 = fmaf(a, Wl7[k * 2 + 0], fmaf(hx, Wr7[k * 2 + 0], z0));
        z1 = fmaf(a, Wl7[k * 2 + 1], fmaf(hx, Wr7[k * 2 + 1], z1));
    }
    float m = fmaxf(z0, z1);
    float lse = m + __logf(__expf(z0 - m) + __expf(z1 - m));
    out[(size_t)n * 2 + 0] = z0 - lse;
    out[(size_t)n * 2 + 1] = z1 - lse;
}

// ---------------------------------------------------------------------------
// Launch: 7 layers. Per layer: init agg -> scatter-max -> dense (WMMA).
// ws layout: h0[N*16] | h1[N*16] | agg[N*16]  (all f32)
// ---------------------------------------------------------------------------
extern "C" void kernel_launch(void* const* d_in, const int* in_sizes, int n_in,
                              void* d_out, int out_size, void* d_ws, size_t ws_size,
                              hipStream_t stream) {
    const float* x      = (const float*)d_in[0];
    const int*   ei     = (const int*)d_in[1];
    const float* Wl1    = (const float*)d_in[2];
    const float* Wr1    = (const float*)d_in[3];
    const float* b1     = (const float*)d_in[4];
    const float* Wl_mid = (const float*)d_in[5];
    const float* Wr_mid = (const float*)d_in[6];
    const float* b_mid  = (const float*)d_in[7];
    const float* Wl7    = (const float*)d_in[8];
    const float* Wr7    = (const float*)d_in[9];
    const float* b7     = (const float*)d_in[10];

    const int N = in_sizes[0] / 16;   // 100000
    const int E = in_sizes[1] / 2;    // 3200000
    const int* src = ei;              // edge_index[0]
    const int* dst = ei + E;          // edge_index[1]

    float* h0  = (float*)d_ws;
    float* h1  = h0 + (size_t)N * 16;
    float* agg = h1 + (size_t)N * 16;

    const int nTiles = N / 16;                    // N is a multiple of 16
    dim3 blk(256);
    dim3 gridFill((N * 16 + 255) / 256);
    dim3 gridEdge((E + 255) / 256);
    dim3 gridDense((nTiles * 32 + 255) / 256);    // 1 wave32 per tile, 8 waves/block
    dim3 gridNode((N + 255) / 256);

    // ---- layer 1: input x -> h0 ----
    fill_neg_inf<<<gridFill, blk, 0, stream>>>((unsigned int*)agg, N * 16);
    sage_scatter_max<<<gridEdge, blk, 0, stream>>>(src, dst, x, agg, E);
    sage_dense_wmma<true><<<gridDense, blk, 0, stream>>>(agg, x, Wl1, Wr1, b1, h0, nTiles);

    // ---- layers 2..6 (5 mid layers), ping-pong h0/h1 ----
    const float* cur = h0;
    float* nxt = h1;
    for (int i = 0; i < 5; ++i) {
        fill_neg_inf<<<gridFill, blk, 0, stream>>>((unsigned int*)agg, N * 16);
        sage_scatter_max<<<gridEdge, blk, 0, stream>>>(src, dst, cur, agg, E);
        sage_dense_wmma<true><<<gridDense, blk, 0, stream>>>(
            agg, cur, Wl_mid + i * 256, Wr_mid + i * 256, b_mid + i * 16, nxt, nTiles);
        const float* t = cur;
        cur = nxt;
        nxt = (float*)t;
    }

    // ---- layer 7: 16 -> 2 + log_softmax ----
    fill_neg_inf<<<gridFill, blk, 0, stream>>>((unsigned int*)agg, N * 16);
    sage_scatter_max<<<gridEdge, blk, 0, stream>>>(src, dst, cur, agg, E);
    sage_final<<<gridNode, blk, 0, stream>>>(agg, cur, Wl7, Wr7, b7, (float*)d_out, N);
}